// DGCNN_44770739093808
// MI455X (gfx1250) — compile-verified
//
#include <hip/hip_runtime.h>

#define EPSBN 1e-5f
#define NEG   0.2f

typedef __attribute__((ext_vector_type(2))) float v2f;
typedef __attribute__((ext_vector_type(8))) float v8f;

// ---------------------------------------------------------------------------
// 1) kNN: one thread per query point, all 2048 points of the batch in LDS.
//    Maintains a sorted top-32 list (order of neighbors is irrelevant: the
//    downstream max over k is permutation invariant).
// ---------------------------------------------------------------------------
__global__ __launch_bounds__(256)
void knn_kernel(const float* __restrict__ points, int* __restrict__ idx, int n)
{
    __shared__ float px[2048], py[2048], pz[2048];
    const int b = blockIdx.z;
    const float* P = points + (size_t)b * 3 * n;
    for (int t = threadIdx.x; t < n; t += blockDim.x) {
        px[t] = P[t]; py[t] = P[n + t]; pz[t] = P[2 * n + t];
    }
    __syncthreads();
    const int i = blockIdx.x * blockDim.x + threadIdx.x;
    if (i >= n) return;
    const float qx = px[i], qy = py[i], qz = pz[i];
    float bd[32]; int bi[32];
    for (int t = 0; t < 32; ++t) { bd[t] = 3.4e38f; bi[t] = 0; }
    for (int j = 0; j < n; ++j) {
        const float dx = qx - px[j], dy = qy - py[j], dz = qz - pz[j];
        const float d = dx * dx + dy * dy + dz * dz;
        if (d < bd[31]) {
            int pos = 31;
            while (pos > 0 && bd[pos - 1] > d) {
                bd[pos] = bd[pos - 1]; bi[pos] = bi[pos - 1]; --pos;
            }
            bd[pos] = d; bi[pos] = j;
        }
    }
    int* out = idx + ((size_t)b * n + i) * 32;
    for (int t = 0; t < 32; ++t) out[t] = bi[t];
}

// ---------------------------------------------------------------------------
// 2) Build Wcat = [ s .* Wd ; s .* (Wc - Wd) ]  (2*cOut x cIn), s = g/sqrt(1+eps)
//    W is (cOut x 2*cIn) row-major:  W[:, :cIn] = Wd, W[:, cIn:] = Wc.
// ---------------------------------------------------------------------------
__global__ void build_wcat(const float* __restrict__ W, const float* __restrict__ g,
                           float* __restrict__ Wcat, int cOut, int cIn)
{
    const int gid = blockIdx.x * blockDim.x + threadIdx.x;
    const int total = 2 * cOut * cIn;
    if (gid >= total) return;
    const int r = gid / cIn, kk = gid - r * cIn;
    const float inv = rsqrtf(1.0f + EPSBN);
    float v;
    if (r < cOut) {
        v = g[r] * inv * W[(size_t)r * (2 * cIn) + kk];
    } else {
        const int ro = r - cOut;
        v = g[ro] * inv * (W[(size_t)ro * (2 * cIn) + cIn + kk] -
                           W[(size_t)ro * (2 * cIn) + kk]);
    }
    Wcat[gid] = v;
}

// ---------------------------------------------------------------------------
// 3) Generic f32 WMMA GEMM:  C[b] = A (MxK, row-major) * B[b] (KxN, generic
//    strides).  Block tile 64(M) x 128(N), K-step 16, 8 waves as 2x4 grid,
//    each wave owns a 32x32 tile = 2x2 accumulators of V_WMMA_F32_16X16X4_F32.
//    Optional fused BN + LeakyReLU epilogue (final layer).
// ---------------------------------------------------------------------------
__global__ __launch_bounds__(256)
void gemm_f32_wmma(const float* __restrict__ A, int lda,
                   const float* __restrict__ B, long bStride, int bRsK, int bRsJ,
                   float* __restrict__ C, long cStride, int cSm, int cSj,
                   int M, int N, int K,
                   const float* __restrict__ gamma, const float* __restrict__ beta,
                   int act)
{
    __shared__ float As[64][17];    // +1 pad: conflict-free strided frag reads
    __shared__ float Bs[16][132];
    const int tid   = threadIdx.x;
    const int lane  = tid & 31, wave = tid >> 5;
    const int waveM = wave >> 2, waveN = wave & 3;   // 2 x 4 wave grid
    const int l15   = lane & 15, half = lane >> 4;
    const int b     = blockIdx.z;
    const float* Bb = B + (size_t)b * bStride;
    float*       Cb = C + (size_t)b * cStride;
    const int gm0 = blockIdx.y * 64;
    const int gj0 = blockIdx.x * 128;

    v8f acc[2][2] = {};
    const int kTiles = (K + 15) >> 4;
    for (int kt = 0; kt < kTiles; ++kt) {
        const int k0 = kt << 4;
        {   // stage A tile (64x16), zero-pad past K
            const int col = tid & 15, row0 = tid >> 4;
            for (int r = 0; r < 4; ++r) {
                const int row = row0 + r * 16;
                const int gk  = k0 + col;
                As[row][col] = (gk < K) ? A[(size_t)(gm0 + row) * lda + gk] : 0.0f;
            }
        }
        {   // stage B tile (16x128); 8 consecutive k per thread (contiguous
            // when the feature matrix is stored channel-contiguous, bRsK==1)
            const int jcol = tid >> 1, ks = (tid & 1) * 8;
            for (int u = 0; u < 8; ++u) {
                const int kk = ks + u;
                const int gk = k0 + kk;
                Bs[kk][jcol] = (gk < K)
                    ? Bb[(size_t)gk * bRsK + (size_t)(gj0 + jcol) * bRsJ] : 0.0f;
            }
        }
        __syncthreads();
        for (int kk = 0; kk < 16; kk += 4) {
            const int kb = kk + half * 2;        // ISA frag layout: lanes 16-31 hold K+2
            v2f a0, a1, b0, b1;
            const int ra = waveM * 32 + l15;
            a0.x = As[ra][kb];        a0.y = As[ra][kb + 1];
            a1.x = As[ra + 16][kb];   a1.y = As[ra + 16][kb + 1];
            const int cb0 = waveN * 32 + l15;
            b0.x = Bs[kb][cb0];       b0.y = Bs[kb + 1][cb0];
            b1.x = Bs[kb][cb0 + 16];  b1.y = Bs[kb + 1][cb0 + 16];
            acc[0][0] = __builtin_amdgcn_wmma_f32_16x16x4_f32(false, a0, false, b0, (short)0, acc[0][0], false, false);
            acc[0][1] = __builtin_amdgcn_wmma_f32_16x16x4_f32(false, a0, false, b1, (short)0, acc[0][1], false, false);
            acc[1][0] = __builtin_amdgcn_wmma_f32_16x16x4_f32(false, a1, false, b0, (short)0, acc[1][0], false, false);
            acc[1][1] = __builtin_amdgcn_wmma_f32_16x16x4_f32(false, a1, false, b1, (short)0, acc[1][1], false, false);
        }
        __syncthreads();
    }
    const float inv = rsqrtf(1.0f + EPSBN);
    for (int mt = 0; mt < 2; ++mt)
        for (int nt = 0; nt < 2; ++nt)
            for (int e = 0; e < 8; ++e) {
                const int m = gm0 + waveM * 32 + mt * 16 + half * 8 + e;
                const int j = gj0 + waveN * 32 + nt * 16 + l15;
                if (m < M && j < N) {
                    float v = acc[mt][nt][e];
                    if (act) {
                        v = gamma[m] * inv * v + beta[m];
                        v = fmaxf(v, NEG * v);   // LeakyReLU(0.2)
                    }
                    Cb[(size_t)m * cSm + (size_t)j * cSj] = v;
                }
            }
}

// ---------------------------------------------------------------------------
// 4) Gather + max over 32 neighbors + folded BN shift + LeakyReLU.
//    HT is (b, n, 2*cOut): row j = [ s*Wd*F_j | s*(Wc-Wd)*F_j ] (contiguous).
//    out = act( max_k HT[idx_k][ch] + HT[i][cOut+ch] + beta[ch] )
// ---------------------------------------------------------------------------
__global__ __launch_bounds__(256)
void gather_max_kernel(const float* __restrict__ HT, const int* __restrict__ idx,
                       const float* __restrict__ beta, float* __restrict__ cat,
                       int cOut, int n, int catOff, int catLd)
{
    const int b   = blockIdx.z;
    const int ppb = blockDim.x / cOut;
    const int p   = threadIdx.x / cOut;
    const int ch  = threadIdx.x - p * cOut;
    const int i   = blockIdx.x * ppb + p;
    const float* HTb = HT + (size_t)b * n * (2 * cOut);
    const int* idxb  = idx + ((size_t)b * n + i) * 32;
    const float tb = HTb[(size_t)i * (2 * cOut) + cOut + ch] + beta[ch];
    float mx = -3.4e38f;
    #pragma unroll 4
    for (int kk = 0; kk < 32; ++kk) {
        const int jn = idxb[kk];
        mx = fmaxf(mx, HTb[(size_t)jn * (2 * cOut) + ch]);  // contiguous across ch
    }
    float y = mx + tb;
    y = fmaxf(y, NEG * y);
    cat[((size_t)b * n + i) * catLd + catOff + ch] = y;
}

// ---------------------------------------------------------------------------
extern "C" void kernel_launch(void* const* d_in, const int* in_sizes, int n_in,
                              void* d_out, int out_size, void* d_ws, size_t ws_size,
                              hipStream_t stream)
{
    const int B = 8, N = 2048, CATC = 512;
    const float* feats = (const float*)d_in[0];
    const float* pts   = (const float*)d_in[1];
    const float* W[4]  = {(const float*)d_in[2], (const float*)d_in[5],
                          (const float*)d_in[8], (const float*)d_in[11]};
    const float* g[4]  = {(const float*)d_in[3], (const float*)d_in[6],
                          (const float*)d_in[9], (const float*)d_in[12]};
    const float* bb[4] = {(const float*)d_in[4], (const float*)d_in[7],
                          (const float*)d_in[10], (const float*)d_in[13]};
    const float* Wout = (const float*)d_in[14];
    const float* gout = (const float*)d_in[15];
    const float* bout = (const float*)d_in[16];

    const int cin [4] = {3, 64, 64, 128};
    const int cout[4] = {64, 64, 128, 256};
    const int coff[4] = {0, 64, 128, 256};

    char* ws = (char*)d_ws;
    size_t off = 0;
    int* d_idx = (int*)(ws + off); off += (size_t)B * N * 32 * sizeof(int);
    float* wcat[4];
    for (int l = 0; l < 4; ++l) {
        wcat[l] = (float*)(ws + off);
        off += (size_t)2 * cout[l] * cin[l] * sizeof(float);
        off = (off + 255) & ~(size_t)255;
    }
    float* dHT  = (float*)(ws + off); off += (size_t)B * N * CATC * sizeof(float);
    float* dcat = (float*)(ws + off); off += (size_t)B * N * CATC * sizeof(float);

    // 1) kNN indices
    knn_kernel<<<dim3(N / 256, 1, B), 256, 0, stream>>>(pts, d_idx, N);

    // 2) folded weight matrices
    for (int l = 0; l < 4; ++l) {
        const int tot = 2 * cout[l] * cin[l];
        build_wcat<<<(tot + 255) / 256, 256, 0, stream>>>(W[l], g[l], wcat[l],
                                                          cout[l], cin[l]);
    }

    // 3) four EdgeConv layers: GEMM (transposed store) + gather-max
    for (int l = 0; l < 4; ++l) {
        const int M = 2 * cout[l], Kd = cin[l];
        const float* Bop; long bStride; int rsK, rsJ;
        if (l == 0) { Bop = feats;               bStride = 3L * N;        rsK = N; rsJ = 1;    }
        else        { Bop = dcat + coff[l - 1];  bStride = (long)N * CATC; rsK = 1; rsJ = CATC; }
        gemm_f32_wmma<<<dim3(N / 128, M / 64, B), 256, 0, stream>>>(
            wcat[l], Kd, Bop, bStride, rsK, rsJ,
            dHT, (long)N * M, /*cSm=*/1, /*cSj=*/M,
            M, N, Kd, nullptr, nullptr, /*act=*/0);
        const int ppb = 256 / cout[l];
        gather_max_kernel<<<dim3(N / ppb, 1, B), 256, 0, stream>>>(
            dHT, d_idx, bb[l], dcat, cout[l], N, coff[l], CATC);
    }

    // 4) final 1024x512 GEMM with fused BN + LeakyReLU, straight into d_out
    gemm_f32_wmma<<<dim3(N / 128, 1024 / 64, B), 256, 0, stream>>>(
        Wout, CATC, dcat, (long)N * CATC, /*rsK=*/1, /*rsJ=*/CATC,
        (float*)d_out, 1024L * N, /*cSm=*/N, /*cSj=*/1,
        1024, N, CATC, gout, bout, /*act=*/1);

    (void)in_sizes; (void)n_in; (void)out_size; (void)ws_size;
}